// GraphCTH_NODE_75780402970749
// MI455X (gfx1250) — compile-verified
//
#include <hip/hip_runtime.h>
#include <math.h>

// Problem constants (match reference)
#define Bq 64
#define Nq 307
#define Tq 24
#define Fq 3
#define Hq 64
#define RQ (Bq * Nq)       // 19648 rows, divisible by 16
#define TAUq 2.0f
#define DTq 0.3f
#define SLOPEq 0.2f
#define EPSq 1e-5f
#define PADW 313           // odd LDS row stride -> conflict-free

typedef __attribute__((ext_vector_type(2))) float v2f;
typedef __attribute__((ext_vector_type(8))) float v8f;

// ---------------------------------------------------------------------------
// Generic fp32 WMMA GEMM: out(rows x 64) = X(rows x K) @ W(K x 64) [+ bias]
// rows = RQ (multiple of 16). 4 waves/block, each wave owns a 16x16 tile.
// X tile staged in LDS (coalesced load, odd stride K+1 -> conflict-free).
// Two accumulators break the WMMA D->C dependency chain.
// ---------------------------------------------------------------------------
template <int K>
__global__ __launch_bounds__(128) void k_gemm64(
    const float* __restrict__ X, const float* __restrict__ W,
    const float* __restrict__ bias, float* __restrict__ out) {
  __shared__ float xt[16 * (K + 1)];
  const int r0 = blockIdx.x * 16;

  // cooperative coalesced load of the 16 x K tile of X
  for (int idx = threadIdx.x; idx < 16 * K; idx += 128) {
    const int rr = idx / K, cc = idx % K;     // K is a power of two
    xt[rr * (K + 1) + cc] = X[(size_t)(r0 + rr) * K + cc];
  }
  __syncthreads();

  const int wave = threadIdx.x >> 5;
  const int lane = threadIdx.x & 31;
  const int m    = lane & 15;
  const int half = lane >> 4;        // 0 or 1
  const int kh   = half * 2;         // K sub-offset per ISA A/B layout
  const int n    = wave * 16 + m;    // output column
  v8f acc0 = {}, acc1 = {};
#pragma unroll
  for (int k = 0; k < K; k += 8) {
    v2f a0, b0, a1, b1;
    a0.x = xt[m * (K + 1) + k + kh];
    a0.y = xt[m * (K + 1) + k + kh + 1];
    b0.x = W[(size_t)(k + kh) * 64 + n];
    b0.y = W[(size_t)(k + kh + 1) * 64 + n];
    a1.x = xt[m * (K + 1) + k + 4 + kh];
    a1.y = xt[m * (K + 1) + k + 4 + kh + 1];
    b1.x = W[(size_t)(k + 4 + kh) * 64 + n];
    b1.y = W[(size_t)(k + 4 + kh + 1) * 64 + n];
    acc0 = __builtin_amdgcn_wmma_f32_16x16x4_f32(false, a0, false, b0,
                                                 (short)0, acc0, false, false);
    acc1 = __builtin_amdgcn_wmma_f32_16x16x4_f32(false, a1, false, b1,
                                                 (short)0, acc1, false, false);
  }
  const float bv = bias ? bias[n] : 0.0f;
#pragma unroll
  for (int i = 0; i < 8; i++) {
    const int mo = i + 8 * half;     // C/D layout: VGPR i -> row i (+8 for hi half)
    out[(size_t)(r0 + mo) * 64 + n] = acc0[i] + acc1[i] + bv;
  }
}

// ---------------------------------------------------------------------------
// es[r] = Wx[r,:] . a_src ; ed[r] = Wx[r,:] . a_dst
// ---------------------------------------------------------------------------
__global__ void k_esed(const float* __restrict__ Wx, const float* __restrict__ asrc,
                       const float* __restrict__ adst, float* __restrict__ es,
                       float* __restrict__ ed) {
  const int r = blockIdx.x * blockDim.x + threadIdx.x;
  if (r >= RQ) return;
  const float* w = Wx + (size_t)r * 64;
  float s = 0.f, d = 0.f;
  for (int c = 0; c < 64; c++) { const float v = w[c]; s += v * asrc[c]; d += v * adst[c]; }
  es[r] = s; ed[r] = d;
}

// ---------------------------------------------------------------------------
// Attention tile: for batch b, node rows [n0, n0+16):
//   e[n,j] = leaky(es[n]+ed[j]) / TAU  masked by A -> softmax over j
//   h[n,:] = tanh( alpha[n,:] @ Wx_b )      (WMMA, K padded to 308)
// ---------------------------------------------------------------------------
__global__ __launch_bounds__(128) void k_attn(
    const float* __restrict__ Wx, const float* __restrict__ es,
    const float* __restrict__ ed, const float* __restrict__ A,
    float* __restrict__ h) {
  __shared__ float al[16 * PADW];
  __shared__ float red[128];
  const int b  = blockIdx.x / 20;
  const int n0 = (blockIdx.x % 20) * 16;
  const float* esb = es + b * Nq;
  const float* edb = ed + b * Nq;
  const float* Wxb = Wx + (size_t)b * Nq * 64;

  const int row = threadIdx.x >> 3;   // 0..15 (8 threads per row)
  const int sub = threadIdx.x & 7;

  // Phase 1: masked leaky-relu logits into LDS (cols 0..307, col 307 = pad)
  {
    const int nn = n0 + row;
    const int nc = (nn < Nq) ? nn : (Nq - 1);   // clamp padded rows (never written out)
    const float esv = esb[nc];
    const float* Arow = A + (size_t)nc * Nq;
    for (int j = sub; j < 308; j += 8) {
      float v;
      if (j < Nq) {
        float e = esv + edb[j];
        e = (e >= 0.f) ? e : SLOPEq * e;
        v = (Arow[j] < 1e-9f) ? -1e30f : (e * (1.0f / TAUq));
      } else {
        v = -1e30f;
      }
      al[row * PADW + j] = v;
    }
  }
  __syncthreads();

  // Phase 2: parallel per-row softmax, 8 threads/row via LDS partials
  float* rowp = al + row * PADW;
  {
    float mx = -1e30f;
    for (int j = sub; j < 308; j += 8) mx = fmaxf(mx, rowp[j]);
    red[threadIdx.x] = mx;
  }
  __syncthreads();
  float rmx = -1e30f;
  for (int i = 0; i < 8; i++) rmx = fmaxf(rmx, red[row * 8 + i]);
  __syncthreads();   // before red[] is reused for sums
  {
    float s = 0.f;
    for (int j = sub; j < 308; j += 8) {
      const float e = __expf(rowp[j] - rmx);
      rowp[j] = e;
      s += e;
    }
    red[threadIdx.x] = s;
  }
  __syncthreads();
  float rs = 0.f;
  for (int i = 0; i < 8; i++) rs += red[row * 8 + i];
  const float inv = 1.0f / rs;
  for (int j = sub; j < 308; j += 8) rowp[j] *= inv;
  __syncthreads();

  // Phase 3: WMMA aggregate alpha(16x308) @ Wx_b(308x64), zero-padded K tail
  const int wave = threadIdx.x >> 5;
  const int lane = threadIdx.x & 31;
  const int m    = lane & 15;
  const int half = lane >> 4;
  const int kh   = half * 2;
  const int n    = wave * 16 + m;
  v8f acc0 = {}, acc1 = {};
  for (int k = 0; k < 304; k += 8) {
    v2f a0, b0, a1, b1;
    a0.x = al[m * PADW + k + kh];
    a0.y = al[m * PADW + k + kh + 1];
    b0.x = Wxb[(size_t)(k + kh) * 64 + n];
    b0.y = Wxb[(size_t)(k + kh + 1) * 64 + n];
    a1.x = al[m * PADW + k + 4 + kh];
    a1.y = al[m * PADW + k + 4 + kh + 1];
    b1.x = Wxb[(size_t)(k + 4 + kh) * 64 + n];
    b1.y = Wxb[(size_t)(k + 4 + kh + 1) * 64 + n];
    acc0 = __builtin_amdgcn_wmma_f32_16x16x4_f32(false, a0, false, b0,
                                                 (short)0, acc0, false, false);
    acc1 = __builtin_amdgcn_wmma_f32_16x16x4_f32(false, a1, false, b1,
                                                 (short)0, acc1, false, false);
  }
  { // tail k = 304..307 (Wx row 307 does not exist -> zero)
    const int k = 304;
    const int kk = k + kh;
    v2f a0, b0;
    a0.x = al[m * PADW + kk];
    a0.y = al[m * PADW + kk + 1];
    b0.x = (kk < Nq)     ? Wxb[(size_t)kk * 64 + n]       : 0.f;
    b0.y = (kk + 1 < Nq) ? Wxb[(size_t)(kk + 1) * 64 + n] : 0.f;
    acc0 = __builtin_amdgcn_wmma_f32_16x16x4_f32(false, a0, false, b0,
                                                 (short)0, acc0, false, false);
  }
#pragma unroll
  for (int i = 0; i < 8; i++) {
    const int mo = i + 8 * half;
    const int nr = n0 + mo;
    if (nr < Nq) h[((size_t)b * Nq + nr) * 64 + n] = tanhf(acc0[i] + acc1[i]);
  }
}

// ---------------------------------------------------------------------------
// out[r,c] = bias[c] + sum_{f<3} x[r*T*F + xoff + f] * W[f*64 + c]
// ---------------------------------------------------------------------------
__global__ void k_proj3(const float* __restrict__ x, int xoff,
                        const float* __restrict__ W, const float* __restrict__ bias,
                        float* __restrict__ out) {
  const int idx = blockIdx.x * blockDim.x + threadIdx.x;
  if (idx >= RQ * 64) return;
  const int r = idx >> 6, c = idx & 63;
  const float* xr = x + (size_t)r * (Tq * Fq) + xoff;
  out[idx] = bias[c] + xr[0] * W[c] + xr[1] * W[64 + c] + xr[2] * W[128 + c];
}

// pred[r, t] = z[r,:] . dec_W + dec_b   (output layout (B,N,T,1))
__global__ void k_pred(const float* __restrict__ z, const float* __restrict__ decW,
                       const float* __restrict__ decb, float* __restrict__ out, int t) {
  const int r = blockIdx.x * blockDim.x + threadIdx.x;
  if (r >= RQ) return;
  const float* zr = z + (size_t)r * 64;
  float s = decb[0];
  for (int c = 0; c < 64; c++) s += zr[c] * decW[c];
  out[(size_t)r * Tq + t] = s;
}

// z[r,:] += DT * layernorm(h[r,:]) * g + b
__global__ void k_ln_update(float* __restrict__ z, const float* __restrict__ h,
                            const float* __restrict__ g, const float* __restrict__ be) {
  const int r = blockIdx.x * blockDim.x + threadIdx.x;
  if (r >= RQ) return;
  const float* hr = h + (size_t)r * 64;
  float m = 0.f;
  for (int c = 0; c < 64; c++) m += hr[c];
  m *= (1.0f / 64.0f);
  float v = 0.f;
  for (int c = 0; c < 64; c++) { const float d = hr[c] - m; v += d * d; }
  v *= (1.0f / 64.0f);
  const float rsq = rsqrtf(v + EPSq);
  float* zr = z + (size_t)r * 64;
  for (int c = 0; c < 64; c++) zr[c] += DTq * ((hr[c] - m) * rsq * g[c] + be[c]);
}

// zcat[r, 0:64] = z ; zcat[r, 64:128] = z_obs
__global__ void k_cat(const float* __restrict__ z, const float* __restrict__ zo,
                      float* __restrict__ zc) {
  const int idx = blockIdx.x * blockDim.x + threadIdx.x;
  if (idx >= RQ * 64) return;
  const int r = idx >> 6, c = idx & 63;
  zc[(size_t)r * 128 + c] = z[idx];
  zc[(size_t)r * 128 + 64 + c] = zo[idx];
}

// z += sigmoid(glin) * (z_obs - z) * mask[r]
__global__ void k_assim(float* __restrict__ z, const float* __restrict__ zo,
                        const float* __restrict__ glin, const float* __restrict__ mask) {
  const int idx = blockIdx.x * blockDim.x + threadIdx.x;
  if (idx >= RQ * 64) return;
  const int r = idx >> 6;
  const float g = 1.0f / (1.0f + __expf(-glin[idx]));
  const float zv = z[idx];
  z[idx] = zv + g * (zo[idx] - zv) * mask[r];
}

extern "C" void kernel_launch(void* const* d_in, const int* in_sizes, int n_in,
                              void* d_out, int out_size, void* d_ws, size_t ws_size,
                              hipStream_t stream) {
  const float* x_seq    = (const float*)d_in[0];
  const float* obs_mask = (const float*)d_in[1];
  const float* A        = (const float*)d_in[2];
  const float* enc_W    = (const float*)d_in[3];
  const float* enc_b    = (const float*)d_in[4];
  const float* g1_W     = (const float*)d_in[5];
  const float* g1_as    = (const float*)d_in[6];
  const float* g1_ad    = (const float*)d_in[7];
  const float* g2_W     = (const float*)d_in[8];
  const float* g2_as    = (const float*)d_in[9];
  const float* g2_ad    = (const float*)d_in[10];
  const float* ln_g     = (const float*)d_in[11];
  const float* ln_b     = (const float*)d_in[12];
  const float* oe_W     = (const float*)d_in[13];
  const float* oe_b     = (const float*)d_in[14];
  const float* gate_W   = (const float*)d_in[15];
  const float* gate_b   = (const float*)d_in[16];
  const float* dec_W    = (const float*)d_in[17];
  const float* dec_b    = (const float*)d_in[18];
  float* out = (float*)d_out;

  // Workspace layout (floats): z | Wx(also gate_lin) | h | z_obs | zcat | es | ed
  float* ws = (float*)d_ws;
  const size_t ZS = (size_t)RQ * 64;
  float* z  = ws;
  float* Wx = z + ZS;
  float* h  = Wx + ZS;
  float* zo = h + ZS;
  float* zc = zo + ZS;                 // RQ * 128
  float* es = zc + (size_t)RQ * 128;
  float* ed = es + RQ;

  const int gemm_grid = RQ / 16;       // 1228
  const int attn_grid = Bq * 20;       // 1280 (ceil(307/16)=20 row tiles)
  const int ew = (RQ * 64 + 255) / 256;
  const int rw = (RQ + 255) / 256;

  // z0 = x_seq[:,:,0,:] @ enc_W + enc_b
  k_proj3<<<ew, 256, 0, stream>>>(x_seq, 0, enc_W, enc_b, z);

  for (int t = 1; t < Tq; t++) {
    // pred from current z (scan emits pred before the update)
    k_pred<<<rw, 256, 0, stream>>>(z, dec_W, dec_b, out, t - 1);
    // GAT layer 1 (tanh fused in k_attn)
    k_gemm64<64><<<gemm_grid, 128, 0, stream>>>(z, g1_W, nullptr, Wx);
    k_esed<<<rw, 256, 0, stream>>>(Wx, g1_as, g1_ad, es, ed);
    k_attn<<<attn_grid, 128, 0, stream>>>(Wx, es, ed, A, h);
    // GAT layer 2
    k_gemm64<64><<<gemm_grid, 128, 0, stream>>>(h, g2_W, nullptr, Wx);
    k_esed<<<rw, 256, 0, stream>>>(Wx, g2_as, g2_ad, es, ed);
    k_attn<<<attn_grid, 128, 0, stream>>>(Wx, es, ed, A, h);
    // z += DT * layernorm(h)
    k_ln_update<<<rw, 256, 0, stream>>>(z, h, ln_g, ln_b);
    // assimilate with x_seq[:,:,t,:]
    k_proj3<<<ew, 256, 0, stream>>>(x_seq, t * Fq, oe_W, oe_b, zo);
    k_cat<<<ew, 256, 0, stream>>>(z, zo, zc);
    k_gemm64<128><<<gemm_grid, 128, 0, stream>>>(zc, gate_W, gate_b, Wx);  // Wx = gate logits
    k_assim<<<ew, 256, 0, stream>>>(z, zo, Wx, obs_mask);
  }
  // final prediction from z_final
  k_pred<<<rw, 256, 0, stream>>>(z, dec_W, dec_b, out, Tq - 1);
}